// SAGEConv_30081950941520
// MI455X (gfx1250) — compile-verified
//
#include <hip/hip_runtime.h>
#include <hip/hip_bf16.h>

typedef __attribute__((ext_vector_type(2))) float v2f;
typedef __attribute__((ext_vector_type(8))) float v8f;

#define C_IN   128
#define C_OUT  128
#define LDA    132   // LDS row pitch in floats: 132 % 64 = 4 -> conflict-free column reads
#define TILE_M 32    // rows per block (two 16-row WMMA tiles per wave, shared B fragments)

// ---------------------------------------------------------------------------
// Kernel 1: zero the workspace (agg + deg). Needed every call (determinism).
// ---------------------------------------------------------------------------
__global__ __launch_bounds__(256) void zero_ws_kernel(float* __restrict__ p, long long n) {
    long long i = (long long)blockIdx.x * blockDim.x + threadIdx.x;
    long long stride = (long long)gridDim.x * blockDim.x;
    for (; i < n; i += stride) p[i] = 0.0f;
}

// ---------------------------------------------------------------------------
// Kernel 2: edge aggregation. One wave per edge, lane = 4-channel chunk.
// Edge index is wave-uniform -> readfirstlane so row/col become scalar loads.
//   agg[row[e], :] += x[col[e], :]   (global_atomic_add_f32, L2-resident)
//   deg[row[e]]    += 1
// ---------------------------------------------------------------------------
__global__ __launch_bounds__(256) void agg_kernel(const float4* __restrict__ x4,
                                                  const int* __restrict__ row,
                                                  const int* __restrict__ col,
                                                  float* __restrict__ agg,
                                                  float* __restrict__ deg,
                                                  int n_edges) {
    const int lane = threadIdx.x & 31;
    int e = __builtin_amdgcn_readfirstlane(blockIdx.x * (blockDim.x >> 5) + (threadIdx.x >> 5));
    const int estride = __builtin_amdgcn_readfirstlane(gridDim.x * (blockDim.x >> 5));
    for (; e < n_edges; e += estride) {
        const int dst = __builtin_amdgcn_readfirstlane(row[e]);   // s_load
        const int src = __builtin_amdgcn_readfirstlane(col[e]);   // s_load
        float4 v = x4[(long long)src * (C_IN / 4) + lane];        // global_load_b128
        float* a = agg + (long long)dst * C_IN + lane * 4;
        atomicAdd(a + 0, v.x);
        atomicAdd(a + 1, v.y);
        atomicAdd(a + 2, v.z);
        atomicAdd(a + 3, v.w);
        if (lane == 0) atomicAdd(&deg[dst], 1.0f);
    }
}

// ---------------------------------------------------------------------------
// Kernel 3: fused  out = (agg/max(deg,1)) @ W  +  x @ Wr  + bias
// Block = 256 threads = 8 waves. Block owns 32 output rows; wave w owns
// output columns [16w, 16w+16) and computes TWO 16x16 tiles, reusing each
// B fragment for both (halves weight fetch per WMMA). fp32 WMMA 16x16x4.
// ---------------------------------------------------------------------------
__global__ __launch_bounds__(256) void sage_gemm_kernel(const float* __restrict__ x,
                                                        const float* __restrict__ W,
                                                        const float* __restrict__ Wr,
                                                        const float* __restrict__ bias,
                                                        const float* __restrict__ agg,
                                                        const float* __restrict__ deg,
                                                        float* __restrict__ out) {
    __shared__ float sM[TILE_M * LDA];   // mean tile, 32 x 128 (padded)
    __shared__ float sX[TILE_M * LDA];   // x    tile, 32 x 128 (padded)

    const int m0  = blockIdx.x * TILE_M;
    const int tid = threadIdx.x;

    // Cooperative load + mean epilogue: 32x128 = 4096 elems, 16 per thread.
    for (int i = tid; i < TILE_M * C_IN; i += 256) {
        int r = i >> 7;            // row in tile
        int c = i & 127;           // channel
        long long node = m0 + r;
        float d  = deg[node];
        float av = agg[node * C_IN + c];
        sM[r * LDA + c] = av / fmaxf(d, 1.0f);
        sX[r * LDA + c] = x[node * C_IN + c];
    }
    __syncthreads();

    const int wave  = tid >> 5;                 // 0..7  -> N-tile
    const int lane  = tid & 31;
    const int mlane = lane & 15;                // A-lane: M%16 ; B-lane: N%16
    const int khalf = (lane >> 4) * 2;          // upper half-wave supplies K+2, K+3
    const int n     = wave * 16 + mlane;        // output column this lane feeds

    v8f acc0 = {};                              // rows [m0,      m0+16)
    v8f acc1 = {};                              // rows [m0+16,   m0+32)

    // mean @ W  (one B fragment feeds both M-tiles)
#pragma unroll
    for (int k = 0; k < C_IN; k += 4) {
        int kk = k + khalf;
        v2f b, a0, a1;
        b.x  = W[(long long)kk * C_OUT + n];
        b.y  = W[(long long)(kk + 1) * C_OUT + n];
        a0.x = sM[mlane * LDA + kk];
        a0.y = sM[mlane * LDA + kk + 1];
        a1.x = sM[(mlane + 16) * LDA + kk];
        a1.y = sM[(mlane + 16) * LDA + kk + 1];
        acc0 = __builtin_amdgcn_wmma_f32_16x16x4_f32(false, a0, false, b,
                                                     (short)0, acc0, false, false);
        acc1 = __builtin_amdgcn_wmma_f32_16x16x4_f32(false, a1, false, b,
                                                     (short)0, acc1, false, false);
    }
    // x @ Wr (accumulate into same tiles)
#pragma unroll
    for (int k = 0; k < C_IN; k += 4) {
        int kk = k + khalf;
        v2f b, a0, a1;
        b.x  = Wr[(long long)kk * C_OUT + n];
        b.y  = Wr[(long long)(kk + 1) * C_OUT + n];
        a0.x = sX[mlane * LDA + kk];
        a0.y = sX[mlane * LDA + kk + 1];
        a1.x = sX[(mlane + 16) * LDA + kk];
        a1.y = sX[(mlane + 16) * LDA + kk + 1];
        acc0 = __builtin_amdgcn_wmma_f32_16x16x4_f32(false, a0, false, b,
                                                     (short)0, acc0, false, false);
        acc1 = __builtin_amdgcn_wmma_f32_16x16x4_f32(false, a1, false, b,
                                                     (short)0, acc1, false, false);
    }

    // Epilogue. C/D layout: VGPR g -> row g (lanes 0-15) / row g+8 (lanes 16-31).
    const float bv    = bias[n];
    const int   mbase = m0 + ((lane >> 4) << 3);
#pragma unroll
    for (int g = 0; g < 8; ++g) {
        out[(long long)(mbase + g) * C_OUT + n]      = acc0[g] + bv;
        out[(long long)(mbase + 16 + g) * C_OUT + n] = acc1[g] + bv;
    }
}

// ---------------------------------------------------------------------------
extern "C" void kernel_launch(void* const* d_in, const int* in_sizes, int n_in,
                              void* d_out, int out_size, void* d_ws, size_t ws_size,
                              hipStream_t stream) {
    const float* x    = (const float*)d_in[0];
    const float* W    = (const float*)d_in[1];
    const float* Wr   = (const float*)d_in[2];
    const float* bias = (const float*)d_in[3];
    const int*   row  = (const int*)d_in[4];
    const int*   col  = (const int*)d_in[5];
    float*       out  = (float*)d_out;

    const int n_nodes = in_sizes[0] / C_IN;
    const int n_edges = in_sizes[4];

    float* agg = (float*)d_ws;                          // [n_nodes, 128]
    float* deg = agg + (long long)n_nodes * C_IN;       // [n_nodes]

    // 1) zero agg + deg
    long long n_zero = (long long)n_nodes * C_IN + n_nodes;
    int zb = (int)((n_zero + 255) / 256);
    if (zb > 65535 * 8) zb = 65535 * 8;
    zero_ws_kernel<<<zb, 256, 0, stream>>>((float*)d_ws, n_zero);

    // 2) edge aggregation: one wave per edge (8 waves per block)
    int ab = (n_edges + 7) / 8;
    agg_kernel<<<ab, 256, 0, stream>>>((const float4*)x, row, col, agg, deg, n_edges);

    // 3) fused mean + dual WMMA GEMM + bias  (n_nodes = 3125 * 32, exact tiles)
    int gb = (n_nodes + TILE_M - 1) / TILE_M;
    sage_gemm_kernel<<<gb, 256, 0, stream>>>(x, W, Wr, bias, agg, deg, out);
}